// SpGraphAttentionLayer_22909355556937
// MI455X (gfx1250) — compile-verified
//
#include <hip/hip_runtime.h>
#include <hip/hip_bf16.h>

typedef __attribute__((ext_vector_type(2))) float v2f;
typedef __attribute__((ext_vector_type(8))) float v8f;

#define GAT_F 128
#define GAT_SLOPE 0.2f

// ---------------- Kernel 1: Wh = h @ W via V_WMMA_F32_16X16X4_F32 ----------------
// Block = 256 threads = 8 waves. Block b covers output rows [16b,16b+16);
// wave w covers cols [16w,16w+16). K-loop of 32 WMMA ops (K=4 each).
__global__ __launch_bounds__(256) void k_gemm_wmma(const float* __restrict__ h,
                                                   const float* __restrict__ Wm,
                                                   float* __restrict__ Wh, int n) {
  const int lane = threadIdx.x & 31;
  const int wtile = threadIdx.x >> 5;       // 0..7 column tile
  const int row0 = blockIdx.x << 4;
  if (row0 >= n) return;                    // uniform: EXEC stays all-ones for WMMA
  const int col0 = wtile << 4;
  const int m = lane & 15;
  const int hi = lane >> 4;                 // 0: K+0..1, 1: K+2..3 (ISA A/B 32-bit layout)

  // A frag: lane(hi,m) holds h[row0+m][k+2*hi], h[row0+m][k+2*hi+1]
  const float* ap = h + (size_t)(row0 + m) * GAT_F + (hi << 1);
  // B frag: lane(hi,m) holds W[k+2*hi][col0+m], W[k+2*hi+1][col0+m]
  const float* bp = Wm + (size_t)(hi << 1) * GAT_F + col0 + m;

  v8f acc = {};
#pragma unroll
  for (int k = 0; k < GAT_F; k += 4) {
    v2f av = *(const v2f*)(ap + k);
    v2f bv;
    bv.x = bp[(size_t)k * GAT_F];
    bv.y = bp[(size_t)k * GAT_F + GAT_F];
    acc = __builtin_amdgcn_wmma_f32_16x16x4_f32(false, av, false, bv, (short)0,
                                                acc, false, false);
  }

  // D layout: VGPR v -> row (row0 + v + 8*hi), col (col0 + m)
  float* op = Wh + (size_t)(row0 + (hi << 3)) * GAT_F + col0 + m;
#pragma unroll
  for (int v = 0; v < 8; ++v) op[(size_t)v * GAT_F] = acc[v];
}

// ---------------- Kernel 2: per-node attention scalars ----------------
// alpha_src[n] = Wh[n] . a[0:128]; alpha_dst[n] = Wh[n] . a[128:256]
__global__ __launch_bounds__(256) void k_alpha(const float* __restrict__ Wh,
                                               const float* __restrict__ a,
                                               float* __restrict__ asrc,
                                               float* __restrict__ adst, int n) {
  const int lane = threadIdx.x & 31;
  const int node = (blockIdx.x * blockDim.x + threadIdx.x) >> 5;
  if (node >= n) return;
  const float* r = Wh + (size_t)node * GAT_F;
  float s0 = 0.f, s1 = 0.f;
#pragma unroll
  for (int i = 0; i < GAT_F; i += 32) {
    float v = r[i + lane];
    s0 += v * a[i + lane];
    s1 += v * a[GAT_F + i + lane];
  }
#pragma unroll
  for (int off = 16; off > 0; off >>= 1) {
    s0 += __shfl_down(s0, off, 32);
    s1 += __shfl_down(s1, off, 32);
  }
  if (lane == 0) { asrc[node] = s0; adst[node] = s1; }
}

// ---------------- Kernel 3: zero accumulators (must run every call) ----------------
__global__ void k_zero(float* __restrict__ out, float* __restrict__ rowsum,
                       int total, int n) {
  const int stride = gridDim.x * blockDim.x;
  const int t0 = blockIdx.x * blockDim.x + threadIdx.x;
  for (int i = t0; i < total; i += stride) out[i] = 0.f;
  for (int i = t0; i < n; i += stride) rowsum[i] = 0.f;
}

// ---------------- Kernel 4: edge weights + rowsum ----------------
__global__ void k_edge_w(const int* __restrict__ row, const int* __restrict__ col,
                         const float* __restrict__ asrc, const float* __restrict__ adst,
                         float* __restrict__ ew, float* __restrict__ rowsum, int e) {
  const int i = blockIdx.x * blockDim.x + threadIdx.x;
  if (i >= e) return;
  const int r = row[i], c = col[i];
  const float s = asrc[r] + adst[c];
  const float lr = s > 0.f ? s : GAT_SLOPE * s;   // leaky_relu(slope=0.2)
  const float w = __expf(-lr);                    // exp(-leaky_relu(score))
  ew[i] = w;
  unsafeAtomicAdd(&rowsum[r], w);
}

// ---------------- Kernel 5: wave-per-edge scatter-add of w * Wh[col] ----------------
__global__ __launch_bounds__(256) void k_aggregate(const int* __restrict__ row,
                                                   const int* __restrict__ col,
                                                   const float* __restrict__ ew,
                                                   const float* __restrict__ Wh,
                                                   float* __restrict__ out, int e) {
  const int edge = (blockIdx.x * blockDim.x + threadIdx.x) >> 5;
  if (edge >= e) return;
  const int lane = threadIdx.x & 31;
  const int r = row[edge], c = col[edge];
  const float w = ew[edge];
  const float4 v = ((const float4*)(Wh + (size_t)c * GAT_F))[lane];  // 512B/wave, L2-resident
  float* dst = out + (size_t)r * GAT_F + (lane << 2);
  unsafeAtomicAdd(dst + 0, w * v.x);
  unsafeAtomicAdd(dst + 1, w * v.y);
  unsafeAtomicAdd(dst + 2, w * v.z);
  unsafeAtomicAdd(dst + 3, w * v.w);
}

// ---------------- Kernel 6: h_prime = elu(acc / rowsum) in-place ----------------
__global__ void k_finalize(float* __restrict__ out, const float* __restrict__ rowsum,
                           int total) {
  const int i = blockIdx.x * blockDim.x + threadIdx.x;
  if (i >= total) return;
  const float x = out[i] / rowsum[i >> 7];        // F=128
  out[i] = x > 0.f ? x : __expf(x) - 1.f;         // elu, alpha=1
}

extern "C" void kernel_launch(void* const* d_in, const int* in_sizes, int n_in,
                              void* d_out, int out_size, void* d_ws, size_t ws_size,
                              hipStream_t stream) {
  const float* h  = (const float*)d_in[0];   // (N,128) f32
  const int*   ei = (const int*)d_in[1];     // (2,E)  i32
  const float* Wm = (const float*)d_in[2];   // (128,128) f32
  const float* a  = (const float*)d_in[3];   // (1,256) f32
  float* out = (float*)d_out;                // (N,128) f32

  const int n = in_sizes[0] / GAT_F;
  const int e = in_sizes[1] / 2;
  const int* row = ei;
  const int* col = ei + e;

  // Workspace carve-out (256B aligned): Wh, asrc, adst, rowsum, ew  (~33 MB)
  char* ws = (char*)d_ws;
  size_t off = 0;
  auto carve = [&](size_t bytes) {
    char* p = ws + off;
    off = (off + bytes + 255) & ~(size_t)255;
    return p;
  };
  float* Wh     = (float*)carve((size_t)n * GAT_F * sizeof(float));
  float* asrc   = (float*)carve((size_t)n * sizeof(float));
  float* adst   = (float*)carve((size_t)n * sizeof(float));
  float* rowsum = (float*)carve((size_t)n * sizeof(float));
  float* ew     = (float*)carve((size_t)e * sizeof(float));
  (void)ws_size; (void)n_in; (void)out_size;

  k_gemm_wmma<<<(n + 15) / 16, 256, 0, stream>>>(h, Wm, Wh, n);
  k_alpha<<<(n + 7) / 8, 256, 0, stream>>>(Wh, a, asrc, adst, n);
  k_zero<<<2048, 256, 0, stream>>>(out, rowsum, n * GAT_F, n);
  k_edge_w<<<(e + 255) / 256, 256, 0, stream>>>(row, col, asrc, adst, ew, rowsum, e);
  k_aggregate<<<(e + 7) / 8, 256, 0, stream>>>(row, col, ew, Wh, out, e);
  k_finalize<<<(n * GAT_F + 255) / 256, 256, 0, stream>>>(out, rowsum, n * GAT_F);
}